// MolecularGNN_Advanced_4105988735687
// MI455X (gfx1250) — compile-verified
//
#include <hip/hip_runtime.h>
#include <hip/hip_bf16.h>

#define N_NODES 50000
#define N_EDGES 600000
#define FH      128
#define NG      2000
#define KZ      384          // 3*H pooled feature width
#define BN_EPS  1e-5f

typedef __attribute__((ext_vector_type(2))) float v2f;
typedef __attribute__((ext_vector_type(8))) float v8f;

static __device__ __forceinline__ void wait_async0() {
#if __has_builtin(__builtin_amdgcn_s_wait_asynccnt)
    __builtin_amdgcn_s_wait_asynccnt(0);
#else
    asm volatile("s_wait_asynccnt 0x0" ::: "memory");
#endif
}

// ---------------- degree / normalization ----------------
__global__ void init_deg_k(float* deg) {
    int i = blockIdx.x * blockDim.x + threadIdx.x;
    if (i < N_NODES) deg[i] = 1.0f;                      // self loop
}

__global__ void deg_accum_k(const int* __restrict__ col, float* deg) {
    int e = blockIdx.x * blockDim.x + threadIdx.x;
    if (e < N_EDGES) atomicAdd(&deg[col[e]], 1.0f);
}

__global__ void dinv_k(float* deg) {
    int i = blockIdx.x * blockDim.x + threadIdx.x;
    if (i < N_NODES) deg[i] = rsqrtf(deg[i]);            // deg >= 1 always
}

// ---------------- weight pre-pack into WMMA B-fragment order ----------------
// Wp[(k/2)*128 + n] = float2{ W[k][n], W[k+1][n] }  -> one b64 load per fragment
__global__ void pack_w_k(const float* __restrict__ W, float* __restrict__ Wp, int Khalf) {
    int idx = blockIdx.x * blockDim.x + threadIdx.x;     // over Khalf*128
    if (idx < Khalf * FH) {
        int kp = idx >> 7, n = idx & (FH - 1);
        Wp[(size_t)idx * 2 + 0] = W[(size_t)(2 * kp)     * FH + n];
        Wp[(size_t)idx * 2 + 1] = W[(size_t)(2 * kp + 1) * FH + n];
    }
}

// ---------------- fp32 WMMA GEMM: out[M,128] = A[M,K] @ W[K,128] (+bias, relu) ----
// Block = 256 threads = 8 waves; block computes a 16-row x 128-col strip.
// A strip staged via async global->LDS (ASYNCcnt path); +4 float LDS padding
// (stride%64==4) keeps the A-fragment ds reads bank-conflict free.
template<int K, int LDK>
__global__ __launch_bounds__(256) void gemm_wmma_k(
    const float* __restrict__ A, const float* __restrict__ Wp,
    const float* __restrict__ bias, float* __restrict__ out, int do_relu) {
    __shared__ float As[16 * LDK];
    const int tid  = threadIdx.x;
    const int row0 = blockIdx.x * 16;
    const int wave = tid >> 5, lane = tid & 31;
    const int half = lane >> 4, q = lane & 15;   // half-wave select / M,N index
    const int n0   = wave * 16;                  // this wave's 16-col tile

    // ---- stage 16 x K A-strip via async copies: one b128 per lane,
    //      one instruction moves a 512B row segment per wave ----
    {
        const unsigned lbase = (unsigned)(uintptr_t)(&As[0]);
        const int segs = K / 128;                // 512B segments per row
        for (int r = wave; r < 16; r += 8) {
            const float* gsrc = A + (size_t)(row0 + r) * K;
            for (int s = 0; s < segs; ++s) {
                unsigned loff = lbase + (unsigned)((r * LDK + s * 128 + lane * 4) * 4);
                const float* g = gsrc + s * 128 + lane * 4;
                asm volatile("global_load_async_to_lds_b128 %0, %1, off"
                             :: "v"(loff), "v"(g) : "memory");
            }
        }
        wait_async0();           // this wave's async copies complete
        __syncthreads();         // all waves' rows visible
    }

    v8f c = {0.f, 0.f, 0.f, 0.f, 0.f, 0.f, 0.f, 0.f};
    for (int kk = 0; kk < K; kk += 4) {
        const int k0 = kk + half * 2;
        v2f a;
        // A 16x4 fragment: lane q holds (M=q, K=k0..k0+1)
        a.x = As[q * LDK + k0];
        a.y = As[q * LDK + k0 + 1];
        // B 4x16 fragment from packed W: single coalesced b64 per lane
        v2f b = *(const v2f*)(Wp + ((size_t)((kk >> 1) + half) * FH + n0 + q) * 2);
        c = __builtin_amdgcn_wmma_f32_16x16x4_f32(
                /*neg_a=*/false, a, /*neg_b=*/false, b,
                /*c_mod=*/(short)0, c, /*reuse_a=*/false, /*reuse_b=*/false);
    }

    const float bv = bias ? bias[n0 + q] : 0.0f;
#pragma unroll
    for (int i = 0; i < 8; ++i) {
        int m = i + half * 8;                    // C/D layout: VGPR i -> M = i (+8 upper half)
        float v = c[i] + bv;
        if (do_relu) v = fmaxf(v, 0.0f);
        out[(size_t)(row0 + m) * FH + n0 + q] = v;
    }
}

// ---------------- self-loop init: agg = tmp * dinv^2 (also zero-fills agg) ----
__global__ void self_loop_k(const float* __restrict__ tmp, const float* __restrict__ dinv,
                            float* __restrict__ agg) {
    int idx = blockIdx.x * blockDim.x + threadIdx.x;
    int n = idx >> 5, lane = idx & 31;
    if (n < N_NODES) {
        float d = dinv[n], w = d * d;
        float4 v = *(const float4*)(tmp + (size_t)n * FH + lane * 4);
        float4 r; r.x = v.x * w; r.y = v.y * w; r.z = v.z * w; r.w = v.w * w;
        *(float4*)(agg + (size_t)n * FH + lane * 4) = r;
    }
}

// ---------------- edge scatter: agg[col] += tmp[row] * dinv[row]*dinv[col] ----
__global__ void scatter_edges_k(const float* __restrict__ tmp, const int* __restrict__ row,
                                const int* __restrict__ col, const float* __restrict__ dinv,
                                float* __restrict__ agg) {
    int idx = blockIdx.x * blockDim.x + threadIdx.x;
    int e = idx >> 5, lane = idx & 31;
    if (e < N_EDGES) {
        int r = row[e], cdst = col[e];
        float w = dinv[r] * dinv[cdst];
        float4 v = *(const float4*)(tmp + (size_t)r * FH + lane * 4);
        float* o = agg + (size_t)cdst * FH + lane * 4;
        atomicAdd(o + 0, v.x * w);
        atomicAdd(o + 1, v.y * w);
        atomicAdd(o + 2, v.z * w);
        atomicAdd(o + 3, v.w * w);
    }
}

// ---------------- fused bias + BatchNorm(eval) + ReLU ----------------
__global__ void bn_relu_k(const float* __restrict__ agg, const float* __restrict__ bias,
                          const float* __restrict__ gamma, const float* __restrict__ beta,
                          const float* __restrict__ mean, const float* __restrict__ var,
                          float* __restrict__ h) {
    int idx = blockIdx.x * blockDim.x + threadIdx.x;
    if (idx < N_NODES * FH) {
        int f = idx & (FH - 1);
        float v = (agg[idx] + bias[f] - mean[f]) * rsqrtf(var[f] + BN_EPS) * gamma[f] + beta[f];
        h[idx] = fmaxf(v, 0.0f);
    }
}

// ---------------- pooling ----------------
__global__ void pool_init_k(float* psum, float* pmax, float* pcnt) {
    int i = blockIdx.x * blockDim.x + threadIdx.x;
    if (i < NG * FH) { psum[i] = 0.0f; pmax[i] = 0.0f; }   // max init 0: post-ReLU h >= 0,
    if (i < NG) pcnt[i] = 0.0f;                            // empty segments -> 0 (matches ref)
}

__global__ void pool_accum_k(const float* __restrict__ h, const int* __restrict__ batch,
                             float* psum, float* pmax, float* pcnt) {
    int idx = blockIdx.x * blockDim.x + threadIdx.x;
    int n = idx >> 5, lane = idx & 31;
    if (n < N_NODES) {
        int g = batch[n];
        float4 v = *(const float4*)(h + (size_t)n * FH + lane * 4);
        float* ps = psum + (size_t)g * FH + lane * 4;
        int*   pm = (int*)(pmax + (size_t)g * FH + lane * 4);
        atomicAdd(ps + 0, v.x); atomicAdd(ps + 1, v.y);
        atomicAdd(ps + 2, v.z); atomicAdd(ps + 3, v.w);
        // non-negative floats: int compare == float compare
        atomicMax(pm + 0, __float_as_int(v.x));
        atomicMax(pm + 1, __float_as_int(v.y));
        atomicMax(pm + 2, __float_as_int(v.z));
        atomicMax(pm + 3, __float_as_int(v.w));
        if (lane == 0) atomicAdd(&pcnt[g], 1.0f);
    }
}

__global__ void build_z_k(const float* __restrict__ psum, const float* __restrict__ pmax,
                          const float* __restrict__ pcnt, float* __restrict__ z) {
    int idx = blockIdx.x * blockDim.x + threadIdx.x;
    if (idx < NG * KZ) {
        int g = idx / KZ, f = idx % KZ;
        float v;
        if (f < FH)            v = psum[g * FH + f] / fmaxf(pcnt[g], 1.0f);  // mean
        else if (f < 2 * FH)   v = pmax[g * FH + (f - FH)];                  // max
        else                   v = psum[g * FH + (f - 2 * FH)];              // sum
        z[idx] = v;
    }
}

// ---------------- final linear: out[g] = zz[g,:] . lin2_W + lin2_b ----------------
__global__ void final_lin_k(const float* __restrict__ zz, const float* __restrict__ w,
                            const float* __restrict__ b, float* __restrict__ out) {
    __shared__ float s[128];
    int g = blockIdx.x, t = threadIdx.x;
    s[t] = zz[(size_t)g * FH + t] * w[t];
    __syncthreads();
    for (int st = 64; st > 0; st >>= 1) {
        if (t < st) s[t] += s[t + st];
        __syncthreads();
    }
    if (t == 0) out[g] = s[0] + b[0];
}

// ---------------- launch ----------------
extern "C" void kernel_launch(void* const* d_in, const int* in_sizes, int n_in,
                              void* d_out, int out_size, void* d_ws, size_t ws_size,
                              hipStream_t stream) {
    (void)in_sizes; (void)n_in; (void)out_size; (void)ws_size;
    const float* x     = (const float*)d_in[0];
    const int*   ei    = (const int*)d_in[1];
    const int*   row   = ei;               // edge_index[0] = sources
    const int*   col   = ei + N_EDGES;     // edge_index[1] = targets
    const int*   batch = (const int*)d_in[2];
    const float* W1 = (const float*)d_in[3];  const float* b1 = (const float*)d_in[4];
    const float* W2 = (const float*)d_in[5];  const float* b2 = (const float*)d_in[6];
    const float* W3 = (const float*)d_in[7];  const float* b3 = (const float*)d_in[8];
    const float* bng = (const float*)d_in[9];
    const float* bnb = (const float*)d_in[10];
    const float* bnm = (const float*)d_in[11];
    const float* bnv = (const float*)d_in[12];
    const float* l1W = (const float*)d_in[13];
    const float* l1b = (const float*)d_in[14];
    const float* l2W = (const float*)d_in[15];
    const float* l2b = (const float*)d_in[16];
    float* out = (float*)d_out;

    // workspace layout (floats): 2 big ping-pong buffers + pools + packed weights
    float* ws   = (float*)d_ws;
    float* dinv = ws;                                   // 50176 (padded)
    float* P    = ws + 50176;                           // [N,128]
    float* Q    = P + (size_t)N_NODES * FH;             // [N,128]
    float* psum = Q + (size_t)N_NODES * FH;             // [G,128]
    float* pmax = psum + (size_t)NG * FH;               // [G,128]
    float* pcnt = pmax + (size_t)NG * FH;               // [G] (padded 2048)
    float* z    = pcnt + 2048;                          // [G,384]
    float* zz   = z + (size_t)NG * KZ;                  // [G,128]
    float* Wp1  = zz + (size_t)NG * FH;                 // [128,128] packed
    float* Wp2  = Wp1 + FH * FH;
    float* Wp3  = Wp2 + FH * FH;
    float* WpZ  = Wp3 + FH * FH;                        // [384,128] packed

    // pack weights into B-fragment order (once per call; cheap)
    pack_w_k<<<(FH / 2 * FH + 255) / 256, 256, 0, stream>>>(W1, Wp1, FH / 2);
    pack_w_k<<<(FH / 2 * FH + 255) / 256, 256, 0, stream>>>(W2, Wp2, FH / 2);
    pack_w_k<<<(FH / 2 * FH + 255) / 256, 256, 0, stream>>>(W3, Wp3, FH / 2);
    pack_w_k<<<(KZ / 2 * FH + 255) / 256, 256, 0, stream>>>(l1W, WpZ, KZ / 2);

    // symmetric normalization D^-1/2 (A+I) D^-1/2
    init_deg_k <<<(N_NODES + 255) / 256, 256, 0, stream>>>(dinv);
    deg_accum_k<<<(N_EDGES + 255) / 256, 256, 0, stream>>>(col, dinv);
    dinv_k     <<<(N_NODES + 255) / 256, 256, 0, stream>>>(dinv);

    auto layer = [&](const float* hin, float* t, float* g, float* hout,
                     const float* Wp, const float* b, int l) {
        gemm_wmma_k<128, 132><<<N_NODES / 16, 256, 0, stream>>>(hin, Wp, nullptr, t, 0);
        self_loop_k    <<<(N_NODES * 32) / 256, 256, 0, stream>>>(t, dinv, g);
        scatter_edges_k<<<(N_EDGES * 32) / 256, 256, 0, stream>>>(t, row, col, dinv, g);
        bn_relu_k      <<<(N_NODES * FH) / 256, 256, 0, stream>>>(
            g, b, bng + l * FH, bnb + l * FH, bnm + l * FH, bnv + l * FH, hout);
    };
    layer(x, P, Q, P, Wp1, b1, 0);
    layer(P, Q, P, Q, Wp2, b2, 1);
    layer(Q, P, Q, P, Wp3, b3, 2);   // final node features in P

    pool_init_k <<<(NG * FH) / 256, 256, 0, stream>>>(psum, pmax, pcnt);
    pool_accum_k<<<(N_NODES * 32) / 256, 256, 0, stream>>>(P, batch, psum, pmax, pcnt);
    build_z_k   <<<(NG * KZ + 255) / 256, 256, 0, stream>>>(psum, pmax, pcnt, z);

    gemm_wmma_k<KZ, KZ + 4><<<NG / 16, 256, 0, stream>>>(z, WpZ, l1b, zz, 1);
    final_lin_k<<<NG, 128, 0, stream>>>(zz, l2W, l2b, out);
}